// MinibatchDiscrimination_79783312490721
// MI455X (gfx1250) — compile-verified
//
#include <hip/hip_runtime.h>
#include <math.h>

// ---------------- problem constants ----------------
#define NB     256          // batch N
#define LDIM   64           // L
#define FDIM   32           // F
#define INF    2048         // L*F
#define OK     1024         // OUT_FEATURES * KERNEL_DIMS
#define OUTF   64
#define KDIM   16
#define OW     (FDIM + OUTF)  // 96 output row width

// ---------------- workspace layout (floats) ----------------
#define WS_T    0            // t = W^T u           (2048)
#define WS_Q    2048         // q = W t             (1024)
#define WS_SCAL 3072         // inv_sigma           (1)
#define WS_M    4096         // m = x W_sn^T + b    (256*1024)

typedef __attribute__((ext_vector_type(2))) float v2f;
typedef __attribute__((ext_vector_type(8))) float v8f;
typedef __attribute__((ext_vector_type(2))) int   i32x2;
typedef __attribute__((ext_vector_type(4))) int   i32x4;

#define AS_G __attribute__((address_space(1)))
#define AS_L __attribute__((address_space(3)))

// ---------------- CDNA5 async global->LDS staging (ASYNCcnt path) ----------------
#if __has_builtin(__builtin_amdgcn_global_load_async_to_lds_b128) && \
    __has_builtin(__builtin_amdgcn_global_load_async_to_lds_b64)
#define USE_ASYNC_LDS 1
#else
#define USE_ASYNC_LDS 0
#endif

__device__ __forceinline__ void async_wait0() {
#if __has_builtin(__builtin_amdgcn_s_wait_asynccnt)
  __builtin_amdgcn_s_wait_asynccnt(0);
#else
  asm volatile("s_wait_asynccnt 0" ::: "memory");
#endif
}

// ============ power iteration: t[c] = sum_r W[r,c]*u[r] ============
__global__ void k_wt_u(const float* __restrict__ W, const float* __restrict__ u,
                       float* __restrict__ t) {
  const int c = blockIdx.x * blockDim.x + threadIdx.x;   // 0..2047
  float acc = 0.f;
  for (int r = 0; r < OK; ++r) acc = fmaf(W[(size_t)r * INF + c], u[r], acc);
  t[c] = acc;
}

// ============ q[r] = sum_c W[r,c]*t[c]  (one wave32 per row) ============
__global__ void k_w_t(const float* __restrict__ W, const float* __restrict__ t,
                      float* __restrict__ q) {
  const int row  = (blockIdx.x * blockDim.x + threadIdx.x) >> 5;  // 0..1023
  const int lane = threadIdx.x & 31;
  const float* wr = W + (size_t)row * INF;
  float acc = 0.f;
  for (int c = lane; c < INF; c += 32) acc = fmaf(wr[c], t[c], acc);
  for (int off = 16; off; off >>= 1) acc += __shfl_xor(acc, off);
  if (lane == 0) q[row] = acc;
}

// ============ inv_sigma = ||t|| / ||q|| ============
__global__ void k_sigma(const float* __restrict__ t, const float* __restrict__ q,
                        float* __restrict__ scal) {
  __shared__ float red[256];
  float a = 0.f, b = 0.f;
  for (int i = threadIdx.x; i < INF; i += 256) { float v = t[i]; a = fmaf(v, v, a); }
  for (int i = threadIdx.x; i < OK;  i += 256) { float v = q[i]; b = fmaf(v, v, b); }
  red[threadIdx.x] = a; __syncthreads();
  for (int s = 128; s; s >>= 1) { if (threadIdx.x < s) red[threadIdx.x] += red[threadIdx.x + s]; __syncthreads(); }
  const float t2 = red[0]; __syncthreads();
  red[threadIdx.x] = b; __syncthreads();
  for (int s = 128; s; s >>= 1) { if (threadIdx.x < s) red[threadIdx.x] += red[threadIdx.x + s]; __syncthreads(); }
  if (threadIdx.x == 0) scal[0] = sqrtf(t2 / red[0]);
}

// ============ m = (x @ W^T) * inv_sigma + b   via V_WMMA_F32_16X16X4_F32 ============
#define KC 32
#define BPAD 36
__global__ void __launch_bounds__(256) k_gemm(const float* __restrict__ X,
                                              const float* __restrict__ W,
                                              const float* __restrict__ bias,
                                              const float* __restrict__ scal,
                                              float* __restrict__ Mout) {
  __shared__ float Al[16 * BPAD];
  __shared__ float Bl[128 * BPAD];

  const int nbase = blockIdx.x * 128;
  const int mbase = blockIdx.y * 16;
  const int tid   = threadIdx.x;
  const int wave  = tid >> 5;
  const int lane  = tid & 31;
  const int half  = lane >> 4;
  const int l15   = lane & 15;
  const int nloc  = wave * 16 + l15;

  const int am = tid >> 4;
  const int ak = (tid & 15) * 2;
  const int bn = tid >> 1;
  const int bk = (tid & 1) * 16;

  v8f c = {0.f, 0.f, 0.f, 0.f, 0.f, 0.f, 0.f, 0.f};

  for (int kb = 0; kb < INF; kb += KC) {
    const float* xs = X + (size_t)(mbase + am) * INF + kb + ak;
    const float* wsrc = W + (size_t)(nbase + bn) * INF + kb + bk;
    float* al = &Al[am * BPAD + ak];
    float* bl = &Bl[bn * BPAD + bk];
#if USE_ASYNC_LDS
    // CDNA5 async DMA: global -> LDS, tracked by ASYNCcnt, no VGPR round-trip
    __builtin_amdgcn_global_load_async_to_lds_b64(
        (AS_G i32x2*)xs, (AS_L i32x2*)al, 0, 0);
    #pragma unroll
    for (int j = 0; j < 4; ++j)
      __builtin_amdgcn_global_load_async_to_lds_b128(
          (AS_G i32x4*)(wsrc + 4 * j), (AS_L i32x4*)(bl + 4 * j), 0, 0);
#else
    al[0] = xs[0]; al[1] = xs[1];
    const float4* wp = (const float4*)wsrc;
    float4* bp = (float4*)bl;
    bp[0] = wp[0]; bp[1] = wp[1]; bp[2] = wp[2]; bp[3] = wp[3];
#endif
    if (kb + KC < INF) {
      __builtin_prefetch(xs + KC, 0, 3);
      __builtin_prefetch(wsrc + KC, 0, 3);
    }
#if USE_ASYNC_LDS
    async_wait0();
#endif
    __syncthreads();

    #pragma unroll
    for (int ks = 0; ks < KC; ks += 4) {
      v2f a, b;
      a[0] = Al[l15 * BPAD + ks + 2 * half];
      a[1] = Al[l15 * BPAD + ks + 2 * half + 1];
      b[0] = Bl[nloc * BPAD + ks + 2 * half];
      b[1] = Bl[nloc * BPAD + ks + 2 * half + 1];
      c = __builtin_amdgcn_wmma_f32_16x16x4_f32(false, a, false, b, (short)0, c,
                                                false, false);
    }
    __syncthreads();
  }

  const float is  = scal[0];
  const int   col = nbase + nloc;
  const float bv  = bias[col];
  #pragma unroll
  for (int r = 0; r < 8; ++r) {
    const int row = mbase + r + 8 * half;
    Mout[(size_t)row * OK + col] = fmaf(c[r], is, bv);
  }
}

// ============ all-pairs L1 + exp ============
__global__ void __launch_bounds__(256) k_pairwise(const float* __restrict__ M,
                                                  float* __restrict__ out) {
  __shared__ float sm[NB * KDIM];
  const int o = blockIdx.x;
  const int i = threadIdx.x;
  const float4* src = (const float4*)(M + (size_t)i * OK + o * KDIM);
  float4 r0 = src[0], r1 = src[1], r2 = src[2], r3 = src[3];
  float4* dst = (float4*)(sm + i * KDIM);
  dst[0] = r0; dst[1] = r1; dst[2] = r2; dst[3] = r3;
  __syncthreads();

  float mi[16] = {r0.x, r0.y, r0.z, r0.w, r1.x, r1.y, r1.z, r1.w,
                  r2.x, r2.y, r2.z, r2.w, r3.x, r3.y, r3.z, r3.w};
  float acc = 0.f;
  for (int j = 0; j < NB; ++j) {
    const float* mj = sm + j * KDIM;
    float s = 0.f;
    #pragma unroll
    for (int k = 0; k < KDIM; ++k) s += fabsf(mi[k] - mj[k]);
    acc += __expf(-s);
  }
  const float ob = acc - 1.f;

  float* op = out + (size_t)i * LDIM * OW + FDIM + o;
  #pragma unroll 4
  for (int l = 0; l < LDIM; ++l) op[(size_t)l * OW] = ob;
}

// ============ copy x into out[:, :, 0:32] ============
__global__ void k_copyx(const float4* __restrict__ x4, float4* __restrict__ out4) {
  const int tid = blockIdx.x * blockDim.x + threadIdx.x;
  const int row = tid >> 3;
  const int f4  = tid & 7;
  out4[(size_t)row * (OW / 4) + f4] = x4[tid];
}

// ---------------- launcher ----------------
extern "C" void kernel_launch(void* const* d_in, const int* in_sizes, int n_in,
                              void* d_out, int out_size, void* d_ws, size_t ws_size,
                              hipStream_t stream) {
  const float* x = (const float*)d_in[0];
  const float* W = (const float*)d_in[1];
  const float* b = (const float*)d_in[2];
  const float* u = (const float*)d_in[3];
  float* out = (float*)d_out;
  float* ws  = (float*)d_ws;

  float* t    = ws + WS_T;
  float* q    = ws + WS_Q;
  float* scal = ws + WS_SCAL;
  float* Mm   = ws + WS_M;

  k_wt_u <<<INF / 256, 256, 0, stream>>>(W, u, t);
  k_w_t  <<<OK / 8,    256, 0, stream>>>(W, t, q);
  k_sigma<<<1,         256, 0, stream>>>(t, q, scal);

  dim3 gg(OK / 128, NB / 16);
  k_gemm<<<gg, 256, 0, stream>>>(x, W, b, scal, Mm);

  k_pairwise<<<OUTF, 256, 0, stream>>>(Mm, out);
  k_copyx<<<(NB * LDIM * FDIM / 4) / 256, 256, 0, stream>>>((const float4*)x,
                                                            (float4*)out);
}